// CRF_17798344474701
// MI455X (gfx1250) — compile-verified
//
#include <hip/hip_runtime.h>

// Linear-chain CRF negative log-likelihood, forward algorithm via f32 WMMA.
// B=256, T=512, N=128.
//
// logsumexp_i(alpha[i]+trans[i,j]) = c + log( sum_i exp(alpha[i]-c) * exp(trans[i,j]) )
// => per-timestep GEMM  S(16x128) = P(16x128) x Etrans(128x128)  per 16-batch tile,
// executed with V_WMMA_F32_16X16X4_F32, Etrans held entirely in registers.
// Software-pipelined: x_{t+1} global loads issued under the GEMM of step t;
// dual WMMA accumulators break the C-operand dependency chain.

#define B_  256
#define T_  512
#define N_  128
#define MROWS 16          // batches per workgroup
#define AST 132           // padded LDS row stride (floats) -> conflict-free
#define NEG_BIG_F  (-1000000000000.0f)
#define MASK_TH    (-1000000.0f)

typedef __attribute__((ext_vector_type(2))) float v2f;
typedef __attribute__((ext_vector_type(8))) float v8f;

__global__ __launch_bounds__(256) void crf_fwd_wmma_kernel(
    const int*   __restrict__ y_true,   // (B,T) int32
    const float* __restrict__ y_pred,   // (B,T,N) f32
    const int*   __restrict__ mask,     // (B,T) int32
    const float* __restrict__ trans,    // (N,N) f32
    float*       __restrict__ out)      // (B,) f32
{
    __shared__ float sAlpha[MROWS * AST];   // alpha tile (16 x 128, padded)
    __shared__ float sP    [MROWS * AST];   // exp(alpha - cmax)
    __shared__ float sX    [MROWS * AST];   // y_pred tile at time t (post mask-add)
    __shared__ float sPmax [MROWS * 16];
    __shared__ float sPmin [MROWS * 16];
    __shared__ float sCmax [MROWS];
    __shared__ float sM    [MROWS];

    const int tid    = threadIdx.x;
    const int row    = tid >> 4;        // 0..15 : batch row within tile
    const int l16    = tid & 15;
    const int c0     = l16 * 8;         // 8 consecutive columns per thread
    const int lane   = tid & 31;        // wave32 lane
    const int hi     = lane >> 4;       // half-wave select
    const int lane15 = lane & 15;
    const int wave   = tid >> 5;        // 8 waves
    const int colbase = wave * 16;      // output column slice of this wave
    const int b0     = blockIdx.x * MROWS;
    const int b      = b0 + row;

    // ---- preload B fragments: Etrans = exp(trans), 128x16 slice per wave, in regs ----
    // B(4x16) layout mirrors documented A(16x4): VGPR0 holds K = 4k + 2*hi,
    // VGPR1 holds K = 4k + 2*hi + 1; col = lane&15.
    v2f bfrag[32];
    {
        const int col = colbase + lane15;
        #pragma unroll
        for (int k = 0; k < 32; ++k) {
            const int r0 = 4 * k + 2 * hi;
            v2f bv;
            bv.x = __expf(trans[r0 * N_ + col]);
            bv.y = __expf(trans[(r0 + 1) * N_ + col]);
            bfrag[k] = bv;
        }
    }

    // ---- software pipeline registers for x_t tile ----
    float xreg[8];
    float addmr = 0.0f;
    auto loadx = [&](int t) {
        const int mk = mask[(size_t)b * T_ + t];
        addmr = (1.0f - (float)mk) * NEG_BIG_F;
        const float* gp = y_pred + ((size_t)b * T_ + t) * N_ + c0;
        #pragma unroll
        for (int i = 0; i < 8; ++i) xreg[i] = gp[i];
        // warm L2 one more timestep ahead (speculative, dropped if invalid)
        __builtin_prefetch((const void*)(gp + N_), 0, 1);
    };

    // E-term accumulators (held by threads tid<16, one per batch row)
    float emit = 0.0f, tsc = 0.0f;
    int   labprev = 0;
    float mprev = 0.0f;

    // ---------------- t = 0 : init alpha = m * yp ----------------
    loadx(0);
    {
        float pmin = 3.4e38f;
        #pragma unroll
        for (int i = 0; i < 8; ++i) {
            const float v = xreg[i] + addmr;
            pmin = fminf(pmin, v);
            sX[row * AST + c0 + i] = v;
        }
        sPmin[row * 16 + l16] = pmin;
        __syncthreads();
        if (tid < MROWS) {
            float mn = 3.4e38f;
            for (int i = 0; i < 16; ++i) mn = fminf(mn, sPmin[tid * 16 + i]);
            const float m = (mn > MASK_TH) ? 1.0f : 0.0f;
            sM[tid] = m;
            const int lab = y_true[(size_t)(b0 + tid) * T_ + 0];
            emit   += m * sX[tid * AST + lab];
            labprev = lab;
            mprev   = m;
        }
        __syncthreads();
        const float m = sM[row];
        #pragma unroll
        for (int i = 0; i < 8; ++i)
            sAlpha[row * AST + c0 + i] = m * sX[row * AST + c0 + i];
        loadx(1);          // issue x_1 loads; latency hides under barrier
        __syncthreads();
    }

    // ---------------- scan t = 1 .. T-1 ----------------
    for (int t = 1; t < T_; ++t) {
        // A: commit prefetched x_t tile to LDS; partial min (mask) + partial max(alpha)
        float pmin = 3.4e38f, pmax = -3.4e38f;
        #pragma unroll
        for (int i = 0; i < 8; ++i) {
            const float v = xreg[i] + addmr;
            pmin = fminf(pmin, v);
            sX[row * AST + c0 + i] = v;
            pmax = fmaxf(pmax, sAlpha[row * AST + c0 + i]);
        }
        sPmin[row * 16 + l16] = pmin;
        sPmax[row * 16 + l16] = pmax;
        __syncthreads();

        // B: per-row reduce cmax + mask flag; fold emit / transition-score terms
        if (tid < MROWS) {
            float mn = 3.4e38f, mx = -3.4e38f;
            for (int i = 0; i < 16; ++i) {
                mn = fminf(mn, sPmin[tid * 16 + i]);
                mx = fmaxf(mx, sPmax[tid * 16 + i]);
            }
            const float m = (mn > MASK_TH) ? 1.0f : 0.0f;
            sM[tid]    = m;
            sCmax[tid] = mx;
            const int lab = y_true[(size_t)(b0 + tid) * T_ + t];
            emit += m * sX[tid * AST + lab];
            tsc  += m * mprev * trans[labprev * N_ + lab];
            labprev = lab;
            mprev   = m;
        }
        __syncthreads();

        // C: P = exp(alpha - cmax); then issue next timestep's global loads
        const float cm = sCmax[row];
        #pragma unroll
        for (int i = 0; i < 8; ++i)
            sP[row * AST + c0 + i] = __expf(sAlpha[row * AST + c0 + i] - cm);
        if (t + 1 < T_) loadx(t + 1);   // latency hides under barrier + GEMM
        __syncthreads();

        // D: GEMM S(16x16) = P(16x128) x Etrans(128x16 slice), K by 4,
        //    two independent accumulators to break the WMMA C-dependency chain
        v8f acc0 = {};
        v8f acc1 = {};
        #pragma unroll
        for (int k = 0; k < 32; k += 2) {
            const int ac0 = 4 * k + 2 * hi;
            const int ac1 = 4 * (k + 1) + 2 * hi;
            v2f a0, a1;
            a0.x = sP[lane15 * AST + ac0];
            a0.y = sP[lane15 * AST + ac0 + 1];
            a1.x = sP[lane15 * AST + ac1];
            a1.y = sP[lane15 * AST + ac1 + 1];
            acc0 = __builtin_amdgcn_wmma_f32_16x16x4_f32(
                false, a0, false, bfrag[k],     (short)0, acc0, false, false);
            acc1 = __builtin_amdgcn_wmma_f32_16x16x4_f32(
                false, a1, false, bfrag[k + 1], (short)0, acc1, false, false);
        }

        // alpha_new = cmax + log(S) + m * x_t   (D layout: row = v + 8*hi, col = lane&15)
        #pragma unroll
        for (int v = 0; v < 8; ++v) {
            const int r  = v + 8 * hi;
            const int cc = colbase + lane15;
            const float s = acc0[v] + acc1[v];
            sAlpha[r * AST + cc] =
                sCmax[r] + __logf(s) + sM[r] * sX[r * AST + cc];
        }
        __syncthreads();
    }

    // ---------------- logZ = logsumexp(alpha) ; out = logZ - (emit + trans_score) ----
    {
        float pmax = -3.4e38f;
        #pragma unroll
        for (int i = 0; i < 8; ++i)
            pmax = fmaxf(pmax, sAlpha[row * AST + c0 + i]);
        sPmax[row * 16 + l16] = pmax;
        __syncthreads();
        if (tid < MROWS) {
            float mx = -3.4e38f;
            for (int i = 0; i < 16; ++i) mx = fmaxf(mx, sPmax[tid * 16 + i]);
            sCmax[tid] = mx;
        }
        __syncthreads();
        const float mx = sCmax[row];
        float s = 0.0f;
        #pragma unroll
        for (int i = 0; i < 8; ++i)
            s += __expf(sAlpha[row * AST + c0 + i] - mx);
        sPmin[row * 16 + l16] = s;
        __syncthreads();
        if (tid < MROWS) {
            float s2 = 0.0f;
            for (int i = 0; i < 16; ++i) s2 += sPmin[tid * 16 + i];
            const float logZ = sCmax[tid] + __logf(s2);
            out[b0 + tid] = logZ - (emit + tsc);
        }
    }
}

extern "C" void kernel_launch(void* const* d_in, const int* in_sizes, int n_in,
                              void* d_out, int out_size, void* d_ws, size_t ws_size,
                              hipStream_t stream) {
    const int*   y_true = (const int*)  d_in[0];   // (B,T) int32
    const float* y_pred = (const float*)d_in[1];   // (B,T,N) f32
    const int*   maskp  = (const int*)  d_in[2];   // (B,T) int32
    const float* trans  = (const float*)d_in[3];   // (N,N) f32
    float* out = (float*)d_out;                    // (B,) f32

    crf_fwd_wmma_kernel<<<B_ / MROWS, 256, 0, stream>>>(y_true, y_pred, maskp, trans, out);
}